// StreamingTransformerLayer_26164940767761
// MI455X (gfx1250) — compile-verified
//
#include <hip/hip_runtime.h>

#define HIDDEN 4096
#define SCALING 2.0f

typedef float v2f __attribute__((ext_vector_type(2)));
typedef float v8f __attribute__((ext_vector_type(8)));

#define NW     8                 // waves per block (wave32 -> 256 threads)
#define KSLAB  (HIDDEN / NW)     // 512 K per wave
#define KC     64                // K-chunk staged in LDS per wave
#define ROWSTR (KC + 4)          // LDS row stride (floats): 68 -> conflict-free b64 reads

// ---------------- gfx1250 async-copy helpers (guarded) ----------------------
__device__ __forceinline__ void async_copy_b128(const float* gsrc, float* ldst) {
#if __has_builtin(__builtin_amdgcn_global_load_async_to_lds_b128)
  typedef int v4i __attribute__((vector_size(16)));
  typedef __attribute__((address_space(1))) v4i glb_v4i;
  typedef __attribute__((address_space(3))) v4i lds_v4i;
  __builtin_amdgcn_global_load_async_to_lds_b128((glb_v4i*)gsrc, (lds_v4i*)ldst, 0, 0);
#else
  *(float4*)ldst = *(const float4*)gsrc;   // synchronous fallback
#endif
}

__device__ __forceinline__ void wait_async() {
#if __has_builtin(__builtin_amdgcn_s_wait_asynccnt)
  __builtin_amdgcn_s_wait_asynccnt(0);
#else
  asm volatile("s_wait_asynccnt 0" ::: "memory");
#endif
  asm volatile("" ::: "memory");   // compiler barrier: keep ds reads after the wait
}

// ---------------- Kernel 1: C4[i][j] = sum_d Bq[d][i] * Av[j][d] ------------
__global__ __launch_bounds__(256) void compute_c4(
    const float* __restrict__ Bq, const float* __restrict__ Av,
    float* __restrict__ C4) {
  __shared__ float s[16];
  const int tid = threadIdx.x;
  if (tid < 16) s[tid] = 0.0f;
  __syncthreads();
  float p[16];
#pragma unroll
  for (int k = 0; k < 16; ++k) p[k] = 0.0f;
  for (int d = tid; d < HIDDEN; d += 256) {
    const float4 bq = *(const float4*)(Bq + (size_t)d * 4);
    const float bqv[4] = {bq.x, bq.y, bq.z, bq.w};
    float avv[4];
#pragma unroll
    for (int j = 0; j < 4; ++j) avv[j] = Av[(size_t)j * HIDDEN + d];
#pragma unroll
    for (int i = 0; i < 4; ++i)
#pragma unroll
      for (int j = 0; j < 4; ++j) p[i * 4 + j] += bqv[i] * avv[j];
  }
#pragma unroll
  for (int k = 0; k < 16; ++k) atomicAdd(&s[k], p[k]);   // ds_add_f32
  __syncthreads();
  if (tid < 16) C4[tid] = s[tid];
}

// ---------------- Kernel 2: low8[row][0..7] = x·[Aq;Av]^T (WMMA) ------------
// Weights are the 16x4 A-matrix (rows 0-3 = Aq, 4-7 = Av, 8-15 = zero),
// 16 x-rows are the 4x16 B-matrix.  Result: lane l<16 holds low8[row l][g]
// in acc[g], lanes 16-31 hold the zero-padded weight rows.
__global__ __launch_bounds__(256) void lora_project(
    const float* __restrict__ x, const float* __restrict__ Aq,
    const float* __restrict__ Av, float* __restrict__ low8) {
  __shared__ float tile[NW][16 * ROWSTR];
  __shared__ float red[16][8];

  const int tid   = threadIdx.x;
  const int wave  = tid >> 5;
  const int lane  = tid & 31;
  const int m     = lane & 15;           // A-matrix row / B-matrix column (x-row)
  const int khalf = (lane >> 4) << 1;    // 0 (lanes 0-15) or 2 (lanes 16-31)
  const int row0  = blockIdx.x << 4;
  const int kslab = wave * KSLAB;

  if (tid < 128) ((float*)red)[tid] = 0.0f;
  __syncthreads();

  const float* wrow = (m < 4) ? (Aq + (size_t)m * HIDDEN)
                    : ((m < 8) ? (Av + (size_t)(m - 4) * HIDDEN) : Aq);
  const float wscale = (m < 8) ? 1.0f : 0.0f;   // zero-pad weight rows 8..15

  float* wtile = tile[wave];
  v8f acc = {0.0f, 0.0f, 0.0f, 0.0f, 0.0f, 0.0f, 0.0f, 0.0f};

  for (int c0 = 0; c0 < KSLAB; c0 += KC) {
    const int kbase = kslab + c0;
    // Stage x[row0..row0+15][kbase..kbase+KC) into this wave's LDS buffer.
    // Per instruction: half-wave = 256 contiguous bytes of one row.
#pragma unroll
    for (int i = 0; i < 8; ++i) {
      const int r = 2 * i + (lane >> 4);
      const int c = (lane & 15) << 2;
      async_copy_b128(x + (size_t)(row0 + r) * HIDDEN + kbase + c,
                      wtile + r * ROWSTR + c);
    }
    wait_async();

#if __has_builtin(__builtin_amdgcn_wmma_f32_16x16x4_f32)
#pragma unroll
    for (int kk = 0; kk < KC; kk += 4) {
      v2f a = *(const v2f*)(wrow + kbase + kk + khalf);   // weights (L0-hot)
      a.x *= wscale; a.y *= wscale;
      v2f b = *(const v2f*)(wtile + m * ROWSTR + kk + khalf);  // x from LDS
      acc = __builtin_amdgcn_wmma_f32_16x16x4_f32(
          /*neg_a=*/false, a, /*neg_b=*/false, b,
          /*c_mod=*/(short)0, acc, /*reuse_a=*/false, /*reuse_b=*/false);
    }
#else
    // Scalar fallback with identical acc layout (lane<16 owns x-row m).
    if (lane < 16) {
      for (int kk = 0; kk < KC; ++kk) {
        const float xv = wtile[m * ROWSTR + kk];
        const int kg = kbase + kk;
#pragma unroll
        for (int g = 0; g < 8; ++g) {
          const float* wr = (g < 4) ? (Aq + (size_t)g * HIDDEN)
                                    : (Av + (size_t)(g - 4) * HIDDEN);
          acc[g] += wr[kg] * xv;
        }
      }
    }
#endif
  }

  // Cross-wave reduction of per-slab partials.
  if (lane < 16) {
#pragma unroll
    for (int g = 0; g < 8; ++g) atomicAdd(&red[lane][g], acc[g]);
  }
  __syncthreads();
  if (tid < 128) {
    const int r = tid >> 3, g = tid & 7;
    low8[((size_t)(row0 + r)) * 8 + g] = red[r][g];
  }
}

// ---------------- Kernel 3: out = 2x + s*lq·Bq^T + s*lv·Bv^T ----------------
__global__ __launch_bounds__(256) void lora_combine(
    const float* __restrict__ x, const float* __restrict__ Bq,
    const float* __restrict__ Bv, const float* __restrict__ low8,
    const float* __restrict__ C4, float* __restrict__ out) {
  const int row = blockIdx.x;
  const size_t base = (size_t)row * HIDDEN;
  const float* lw = low8 + (size_t)row * 8;

  float lq[4], cq[4], cv[4];
#pragma unroll
  for (int r = 0; r < 4; ++r) lq[r] = lw[r];
#pragma unroll
  for (int r = 0; r < 4; ++r) {
    const float lv = lw[4 + r] + SCALING * (lq[0] * C4[0 * 4 + r] +
                                            lq[1] * C4[1 * 4 + r] +
                                            lq[2] * C4[2 * 4 + r] +
                                            lq[3] * C4[3 * 4 + r]);
    cq[r] = SCALING * lq[r];
    cv[r] = SCALING * lv;
  }

#pragma unroll
  for (int i = 0; i < 4; ++i) {
    const int d0 = (threadIdx.x << 2) + (i << 10);
    const float4 xv = *(const float4*)(x + base + d0);
    const float xa[4] = {xv.x, xv.y, xv.z, xv.w};
    float o[4];
#pragma unroll
    for (int c = 0; c < 4; ++c) {
      const int d = d0 + c;
      const float4 bq = *(const float4*)(Bq + (size_t)d * 4);  // L2/L0-resident
      const float4 bv = *(const float4*)(Bv + (size_t)d * 4);
      o[c] = 2.0f * xa[c]
           + cq[0] * bq.x + cq[1] * bq.y + cq[2] * bq.z + cq[3] * bq.w
           + cv[0] * bv.x + cv[1] * bv.y + cv[2] * bv.z + cv[3] * bv.w;
    }
    *(float4*)(out + base + d0) = make_float4(o[0], o[1], o[2], o[3]);
  }
}

// ----------------------------------------------------------------------------
extern "C" void kernel_launch(void* const* d_in, const int* in_sizes, int n_in,
                              void* d_out, int out_size, void* d_ws, size_t ws_size,
                              hipStream_t stream) {
  (void)n_in; (void)out_size; (void)ws_size;
  const float* x  = (const float*)d_in[0];
  const float* Aq = (const float*)d_in[1];   // [4, 4096]
  const float* Bq = (const float*)d_in[2];   // [4096, 4]
  const float* Av = (const float*)d_in[3];   // [4, 4096]
  const float* Bv = (const float*)d_in[4];   // [4096, 4]
  float* out = (float*)d_out;
  const int nrows = in_sizes[0] / HIDDEN;    // 8192

  float* C4   = (float*)d_ws;                // 16 floats
  float* low8 = (float*)d_ws + 64;           // nrows * 8 floats (256 KB)

  compute_c4<<<1, 256, 0, stream>>>(Bq, Av, C4);
  lora_project<<<nrows / 16, 256, 0, stream>>>(x, Aq, Av, low8);
  lora_combine<<<nrows, 256, 0, stream>>>(x, Bq, Bv, low8, C4, out);
}